// KWDSConv2D_58712202936583
// MI455X (gfx1250) — compile-verified
//
#include <hip/hip_runtime.h>
#include <hip/hip_bf16.h>

typedef __attribute__((ext_vector_type(2))) float v2f;
typedef __attribute__((ext_vector_type(8))) float v8f;

#define B_    8
#define C_    128
#define H_    256
#define W_    256
#define HW_   (H_ * W_)
#define NK_   4
#define OC_   128
#define TN    64            // pixels per tile (one row segment)
#define PSTR  130           // floats per LDS row-pair: 2*TN + 2 (8B aligned, conflict-light)

// ---------------- Kernel 1: global average pool  x[B,C,H,W] -> pooled[B*C] ----
__global__ __launch_bounds__(256) void kwds_pool(const float* __restrict__ x,
                                                 float* __restrict__ pooled) {
    const int bc = blockIdx.x;                       // 0 .. B*C-1
    const float* p = x + (size_t)bc * HW_;
    float s = 0.f;
    for (int i = threadIdx.x; i < HW_; i += 256) s += p[i];
    __shared__ float red[256];
    red[threadIdx.x] = s;
    __syncthreads();
    for (int off = 128; off > 0; off >>= 1) {
        if (threadIdx.x < off) red[threadIdx.x] += red[threadIdx.x + off];
        __syncthreads();
    }
    if (threadIdx.x == 0) pooled[bc] = red[0] * (1.0f / (float)HW_);
}

// ------------- Kernel 2: attention softmax + dynamic depthwise kernel --------
// dyn[b][c][9] = sum_k softmax(pooled[b]@attn_w.T + attn_b)[k] * bank[k][c][9]
__global__ __launch_bounds__(128) void kwds_attn(const float* __restrict__ pooled,
                                                 const float* __restrict__ attn_w,
                                                 const float* __restrict__ attn_b,
                                                 const float* __restrict__ bank,
                                                 float* __restrict__ dyn) {
    const int b = blockIdx.x;
    const int t = threadIdx.x;                       // 0..127 == channel
    __shared__ float red[128];
    __shared__ float attn_s[NK_];
    const float pv = pooled[b * C_ + t];
    float logits[NK_];
    for (int k = 0; k < NK_; ++k) {
        red[t] = pv * attn_w[k * C_ + t];
        __syncthreads();
        for (int off = 64; off > 0; off >>= 1) {
            if (t < off) red[t] += red[t + off];
            __syncthreads();
        }
        logits[k] = red[0] + attn_b[k];
        __syncthreads();
    }
    if (t == 0) {
        float mx = logits[0];
        for (int k = 1; k < NK_; ++k) mx = fmaxf(mx, logits[k]);
        float e[NK_], sum = 0.f;
        for (int k = 0; k < NK_; ++k) { e[k] = __expf(logits[k] - mx); sum += e[k]; }
        const float inv = 1.0f / sum;
        for (int k = 0; k < NK_; ++k) attn_s[k] = e[k] * inv;
    }
    __syncthreads();
    const float a0 = attn_s[0], a1 = attn_s[1], a2 = attn_s[2], a3 = attn_s[3];
    #pragma unroll
    for (int idx = 0; idx < 9; ++idx) {
        dyn[(b * C_ + t) * 9 + idx] =
            a0 * bank[(0 * C_ + t) * 9 + idx] +
            a1 * bank[(1 * C_ + t) * 9 + idx] +
            a2 * bank[(2 * C_ + t) * 9 + idx] +
            a3 * bank[(3 * C_ + t) * 9 + idx];
    }
}

// -------- Kernel 3: fused depthwise-3x3 (LDS) + pointwise GEMM via WMMA ------
// Block = (batch b, 64-pixel row segment). dw staged in LDS with a row-pair
// interleaved layout so each WMMA B-fragment {dw[c][n], dw[c+1][n]} is one
// 8-byte-contiguous ds_load_b64 straight into an even-aligned VGPR pair.
// Then out[128][64] = pw[128x128] * dw[128x64] via v_wmma_f32_16x16x4_f32.
__global__ __launch_bounds__(256) void kwds_fused(const float* __restrict__ x,
                                                  const float* __restrict__ dyn,
                                                  const float* __restrict__ pw,
                                                  const float* __restrict__ pwb,
                                                  float* __restrict__ out) {
    __shared__ float kern[C_ * 9];                   // 4.5 KB
    __shared__ float dw2[(C_ / 2) * PSTR];           // 32.5 KB: [c/2][n*2 + (c&1)]

    const int blk  = blockIdx.x;                     // 0 .. 8191
    const int b    = blk >> 10;                      // / (HW_/TN)
    const int tile = blk & 1023;
    const int p0   = tile * TN;                      // first pixel of tile
    const int h    = p0 >> 8;                        // row (W_ == 256)
    const int w0   = p0 & (W_ - 1);
    const int t    = threadIdx.x;

    // stage this batch's dynamic depthwise kernel
    for (int i = t; i < C_ * 9; i += 256) kern[i] = dyn[b * C_ * 9 + i];
    __syncthreads();

    // ---- depthwise 3x3 into LDS: thread (cg, n) covers c = cg*32 + it ----
    const int n  = t & 63;                           // pixel within tile
    const int cg = t >> 6;                           // 0..3 channel group
    const int w  = w0 + n;
    const float* xb = x + (size_t)b * C_ * HW_;
    for (int it = 0; it < 32; ++it) {
        const int c = cg * 32 + it;
        const float* kc = &kern[c * 9];
        const float* xc = xb + (size_t)c * HW_;
        if (it + 1 < 32)                              // gfx1250 global_prefetch_b8
            __builtin_prefetch(xb + (size_t)(c + 1) * HW_ + h * W_ + w, 0, 0);
        float acc = 0.f;
        #pragma unroll
        for (int i = 0; i < 3; ++i) {
            const int hh = h + i - 1;
            if (hh >= 0 && hh < H_) {
                const float* row = xc + hh * W_;
                #pragma unroll
                for (int j = 0; j < 3; ++j) {
                    const int ww = w + j - 1;
                    if (ww >= 0 && ww < W_) acc += kc[i * 3 + j] * row[ww];
                }
            }
        }
        // row-pair interleave: element (c, n) -> dw2[c>>1][2n + (c&1)]
        dw2[(c >> 1) * PSTR + 2 * n + (c & 1)] = acc;
    }
    __syncthreads();

    // ---- pointwise GEMM: 8 waves, each a 16(M) x 64(N) strip ----
    const int wave  = t >> 5;
    const int lane  = t & 31;
    const int l16   = lane & 15;
    const int hi    = lane >> 4;                     // 0: K rows {0,1}, 1: {2,3}
    const int mbase = wave * 16;
    const int m     = mbase + l16;
    const int kb    = hi * 2;

    v8f acc0 = {}, acc1 = {}, acc2 = {}, acc3 = {};
    #pragma unroll
    for (int kk = 0; kk < 32; ++kk) {
        // A-frag per ISA 16x4 f32 layout: lanes 0-15 hold K {0,1}, 16-31 {2,3}
        // contiguous pair -> single global_load_b64
        const v2f a = *(const v2f*)&pw[m * C_ + kk * 4 + kb];
        // B-frags: pair index (4kk+2hi)/2 = 2kk+hi, one ds_load_b64 each
        const float* bp = &dw2[(2 * kk + hi) * PSTR];
        const v2f b0 = *(const v2f*)&bp[2 * (0 * 16 + l16)];
        const v2f b1 = *(const v2f*)&bp[2 * (1 * 16 + l16)];
        const v2f b2 = *(const v2f*)&bp[2 * (2 * 16 + l16)];
        const v2f b3 = *(const v2f*)&bp[2 * (3 * 16 + l16)];
        acc0 = __builtin_amdgcn_wmma_f32_16x16x4_f32(false, a, false, b0, (short)0, acc0, false, false);
        acc1 = __builtin_amdgcn_wmma_f32_16x16x4_f32(false, a, false, b1, (short)0, acc1, false, false);
        acc2 = __builtin_amdgcn_wmma_f32_16x16x4_f32(false, a, false, b2, (short)0, acc2, false, false);
        acc3 = __builtin_amdgcn_wmma_f32_16x16x4_f32(false, a, false, b3, (short)0, acc3, false, false);
    }

    // D layout: VGPR r -> M = mbase + r (lanes 0-15) / mbase + 8 + r (lanes 16-31)
    float* outb = out + (size_t)b * OC_ * HW_;
    #pragma unroll
    for (int r = 0; r < 8; ++r) {
        const int mm   = mbase + hi * 8 + r;
        float* orow    = outb + (size_t)mm * HW_ + p0;
        const float bb = pwb[mm];
        orow[ 0 + l16] = acc0[r] + bb;
        orow[16 + l16] = acc1[r] + bb;
        orow[32 + l16] = acc2[r] + bb;
        orow[48 + l16] = acc3[r] + bb;
    }
}

extern "C" void kernel_launch(void* const* d_in, const int* in_sizes, int n_in,
                              void* d_out, int out_size, void* d_ws, size_t ws_size,
                              hipStream_t stream) {
    const float* x      = (const float*)d_in[0];   // [8,128,256,256]
    const float* bank   = (const float*)d_in[1];   // [4,128,1,3,3]
    const float* attn_w = (const float*)d_in[2];   // [4,128]
    const float* attn_b = (const float*)d_in[3];   // [4]
    const float* pw_w   = (const float*)d_in[4];   // [128,128,1,1]
    const float* pw_b   = (const float*)d_in[5];   // [128]
    float* out = (float*)d_out;                    // [8,128,256,256]

    float* pooled = (float*)d_ws;                  // [B*C] = 1024 floats
    float* dyn    = pooled + B_ * C_;              // [B*C*9] = 9216 floats

    kwds_pool<<<B_ * C_, 256, 0, stream>>>(x, pooled);
    kwds_attn<<<B_, 128, 0, stream>>>(pooled, attn_w, attn_b, bank, dyn);
    kwds_fused<<<B_ * (HW_ / TN), 256, 0, stream>>>(x, dyn, pw_w, pw_b, out);
}